// DMTetGeometry_79182017069559
// MI455X (gfx1250) — compile-verified
//
#include <hip/hip_runtime.h>
#include <hip/hip_bf16.h>
#include <math.h>

// ---------------------------------------------------------------------------
// Marching tetrahedra (DMTet) pipeline for gfx1250 (MI455X).
//   inputs : pos [NV,3] f32, sdf [NV] f32, tet [NT,4] i32
//   outputs (flattened into d_out as f32, reference return order):
//     verts [Ne,3] | faces [Nf,3] | uvs [N*N*4,2] | uv_idx [Nf,3]
//
// Design notes (MI455X):
//  * All hot data (tet 12.8MB, sdf 0.8MB, pos 2.4MB, hash 50MB) is L2-resident
//    (192MB), so random gathers and hash probes are L2-latency, not HBM-bound.
//  * Total traffic ~150-200MB -> ~10us at 23.3TB/s; the bound is hash-CAS
//    latency, so the pipeline minimizes full passes (6 kernels, no scan-fixup
//    pass: block offsets folded into the consumer).
//  * CDNA5-specific paths: global_load_async_to_lds_b32 (ASYNCcnt) stages the
//    triangle tables into LDS; global_prefetch_b8 warms the pos gathers.
//    No WMMA: this workload has no dense matrix structure.
// ---------------------------------------------------------------------------

#define HASH_LOG   22
#define HASH_SIZE  (1u << HASH_LOG)
#define HMASK      (HASH_SIZE - 1u)
#define EMPTY_KEY  (~0ull)
#define MAXPROBE   8192
#define SCAN_BS    256

// Packed tables: [0..95] = TRIANGLE_TABLE (16x6), [96..111] = NUM_TRI_TABLE
__constant__ int d_packed[112] = {
    -1,-1,-1,-1,-1,-1,   1, 0, 2,-1,-1,-1,   4, 0, 3,-1,-1,-1,   1, 4, 2, 1, 3, 4,
     3, 1, 5,-1,-1,-1,   2, 3, 0, 2, 5, 3,   1, 4, 0, 1, 5, 4,   4, 2, 5,-1,-1,-1,
     4, 5, 2,-1,-1,-1,   4, 1, 0, 4, 5, 1,   3, 2, 0, 3, 5, 2,   1, 3, 5,-1,-1,-1,
     4, 1, 2, 4, 3, 1,   3, 0, 4,-1,-1,-1,   2, 0, 1,-1,-1,-1,  -1,-1,-1,-1,-1,-1,
    // NUM_TRI_TABLE
     0, 1, 1, 2, 1, 2, 2, 1, 1, 2, 2, 1, 2, 1, 1, 0
};
__constant__ int d_numtri[16] = {0,1,1,2,1,2,2,1,1,2,2,1,2,1,1,0};
__constant__ int d_ea[6] = {0,0,0,1,1,2};
__constant__ int d_eb[6] = {1,2,3,2,3,3};

__device__ __forceinline__ uint32_t hash_key(unsigned long long k) {
    k ^= k >> 33; k *= 0xff51afd7ed558ccdULL;
    k ^= k >> 33; k *= 0xc4ceb9fe1a85ec53ULL;
    k ^= k >> 33;
    return (uint32_t)k & HMASK;
}

// --------------------------------------------------------------- kernel: init
// 2 x u64 per lane -> b128-wide stores over the 33MB key table.
__global__ void k_init(ulonglong2* __restrict__ hkeys2,
                       uint32_t* __restrict__ counters) {
    uint32_t i = blockIdx.x * blockDim.x + threadIdx.x;
    if (i < (HASH_SIZE / 2u)) hkeys2[i] = make_ulonglong2(EMPTY_KEY, EMPTY_KEY);
    if (i < 16u)              counters[i] = 0u;
}

// --------------------------------------- kernel: classify + block scan (pass1)
// scan value packs (#tets with 1 tri) in low 32, (#tets with 2 tris) in high 32
__global__ void k_classify_scan1(const int* __restrict__ tet,
                                 const float* __restrict__ sdf, int NT,
                                 uint32_t* __restrict__ code,
                                 unsigned long long* __restrict__ scanArr,
                                 unsigned long long* __restrict__ blockSums) {
    __shared__ unsigned long long sh[SCAN_BS];
    int t = blockIdx.x * SCAN_BS + threadIdx.x;
    unsigned long long v = 0ull;
    uint32_t c = 0xFFu;
    if (t < NT) {
        int4 vv = ((const int4*)tet)[t];
        int b0 = sdf[vv.x] > 0.f, b1 = sdf[vv.y] > 0.f;
        int b2 = sdf[vv.z] > 0.f, b3 = sdf[vv.w] > 0.f;
        uint32_t cc = (uint32_t)(b0 | (b1 << 1) | (b2 << 2) | (b3 << 3));
        if (cc != 0u && cc != 15u) {
            c = cc;
            int ntri = d_numtri[cc];
            v = (ntri == 1) ? 1ull : (ntri == 2 ? (1ull << 32) : 0ull);
        }
        code[t] = c;
    }
    sh[threadIdx.x] = v;
    __syncthreads();
    for (int off = 1; off < SCAN_BS; off <<= 1) {
        unsigned long long add = (threadIdx.x >= (unsigned)off) ? sh[threadIdx.x - off] : 0ull;
        __syncthreads();
        sh[threadIdx.x] += add;
        __syncthreads();
    }
    unsigned long long incl = sh[threadIdx.x];
    if (t < NT) scanArr[t] = incl - v;                 // exclusive within block
    if (threadIdx.x == SCAN_BS - 1) blockSums[blockIdx.x] = incl;
}

// ------------------------------------------------ kernel: scan of block sums
__global__ void k_scan2(unsigned long long* __restrict__ blockSums, int nb,
                        uint32_t* __restrict__ counters) {
    __shared__ unsigned long long sh[1024];
    __shared__ unsigned long long carry;
    if (threadIdx.x == 0) carry = 0ull;
    __syncthreads();
    for (int base = 0; base < nb; base += 1024) {
        int i = base + (int)threadIdx.x;
        unsigned long long v = (i < nb) ? blockSums[i] : 0ull;
        sh[threadIdx.x] = v;
        __syncthreads();
        for (int off = 1; off < 1024; off <<= 1) {
            unsigned long long add = (threadIdx.x >= (unsigned)off) ? sh[threadIdx.x - off] : 0ull;
            __syncthreads();
            sh[threadIdx.x] += add;
            __syncthreads();
        }
        unsigned long long incl = sh[threadIdx.x];
        if (i < nb) blockSums[i] = carry + incl - v;   // exclusive, global
        __syncthreads();
        if (threadIdx.x == 1023) carry += sh[1023];
        __syncthreads();
    }
    if (threadIdx.x == 0) {
        counters[1] = (uint32_t)(carry & 0xffffffffull);  // total #1-tri tets
        counters[2] = (uint32_t)(carry >> 32);            // total #2-tri tets
    }
}

// ------------------------- kernel: edge dedup (hash CAS) + vertex interpolate
__global__ void k_edges(const int* __restrict__ tet, const float* __restrict__ sdf,
                        const float* __restrict__ pos, int NT,
                        const uint32_t* __restrict__ code,
                        unsigned long long* __restrict__ hkeys,
                        uint32_t* __restrict__ hval,
                        uint32_t* __restrict__ counters,
                        float* __restrict__ out /* verts at offset 0 */) {
    int t = blockIdx.x * blockDim.x + threadIdx.x;
    if (t >= NT) return;
    uint32_t c = code[t];
    if (c == 0xFFu) return;
    int4 vv = ((const int4*)tet)[t];
    int vid[4] = {vv.x, vv.y, vv.z, vv.w};
    #pragma unroll
    for (int e = 0; e < 6; ++e) {
        int ia = d_ea[e], ib = d_eb[e];
        uint32_t oa = (c >> ia) & 1u, ob = (c >> ib) & 1u;
        if (oa == ob) continue;                         // not sign-crossing
        int a = vid[ia], b = vid[ib];
        int lo = (a < b) ? a : b;
        int hi = (a < b) ? b : a;
        // gfx1250 prefetch of gather targets (L2-resident; warms WGP$)
        __builtin_prefetch(&pos[3 * (size_t)lo], 0, 0);
        __builtin_prefetch(&pos[3 * (size_t)hi], 0, 0);
        unsigned long long key =
            ((unsigned long long)(uint32_t)lo << 32) | (unsigned long long)(uint32_t)hi;
        uint32_t h = hash_key(key);
        for (int p = 0; p < MAXPROBE; ++p, h = (h + 1u) & HMASK) {
            unsigned long long old = atomicCAS(&hkeys[h], EMPTY_KEY, key);
            if (old == EMPTY_KEY) {
                uint32_t vi = atomicAdd(&counters[0], 1u);
                hval[h] = vi;
                float sa = sdf[lo], sb = sdf[hi];
                float denom = sa - sb;                  // sa>0 xor sb>0 -> != 0
                float wlo = -sb / denom, whi = sa / denom;
                const float* pa = &pos[3 * (size_t)lo];
                const float* pb = &pos[3 * (size_t)hi];
                float* dst = &out[3 * (size_t)vi];
                dst[0] = pa[0] * wlo + pb[0] * whi;
                dst[1] = pa[1] * wlo + pb[1] * whi;
                dst[2] = pa[2] * wlo + pb[2] * whi;
                break;
            }
            if (old == key) break;                      // already owned
        }
    }
}

__device__ __forceinline__ int lookup_edge(const unsigned long long* __restrict__ hkeys,
                                           const uint32_t* __restrict__ hval,
                                           unsigned long long key) {
    uint32_t h = hash_key(key);
    for (int p = 0; p < MAXPROBE; ++p, h = (h + 1u) & HMASK) {
        unsigned long long k = hkeys[h];
        if (k == key) return (int)hval[h];
        if (k == EMPTY_KEY) return -1;
    }
    return -1;
}

// ---------------------------------------- kernel: emit faces + uv_idx
// Same block partition as k_classify_scan1: block offset (blockSums[blockIdx.x])
// is folded in here, eliminating the separate scan-fixup pass.
__global__ void k_faces(const int* __restrict__ tet, int NT, int NN,
                        const uint32_t* __restrict__ code,
                        const unsigned long long* __restrict__ scanArr,
                        const unsigned long long* __restrict__ blockSums,
                        const unsigned long long* __restrict__ hkeys,
                        const uint32_t* __restrict__ hval,
                        const uint32_t* __restrict__ counters,
                        float* __restrict__ out) {
    __shared__ int shTbl[112];
    // --- CDNA5 async copy unit: stage tri/numtri tables global -> LDS -------
    {
        int lane = (int)threadIdx.x;
        if (lane < 112) {
            const int* src = &d_packed[0];
            uint32_t ldsAddr = (uint32_t)(uintptr_t)(&shTbl[lane]);
            uint32_t voff = (uint32_t)lane * 4u;
            asm volatile("global_load_async_to_lds_b32 %0, %1, %2 offset:0"
                         :: "v"(ldsAddr), "v"(voff), "s"(src) : "memory");
        }
        asm volatile("s_wait_asynccnt 0" ::: "memory");
        __syncthreads();
    }

    int t = blockIdx.x * blockDim.x + threadIdx.x;
    if (t >= NT) return;
    uint32_t c = code[t];
    if (c == 0xFFu) return;
    int ntri = shTbl[96 + (int)c];
    if (ntri == 0) return;

    int4 vv = ((const int4*)tet)[t];
    int vid[4] = {vv.x, vv.y, vv.z, vv.w};
    int evid[6];
    #pragma unroll
    for (int e = 0; e < 6; ++e) {
        int ia = d_ea[e], ib = d_eb[e];
        uint32_t oa = (c >> ia) & 1u, ob = (c >> ib) & 1u;
        evid[e] = -1;
        if (oa != ob) {
            int a = vid[ia], b = vid[ib];
            int lo = (a < b) ? a : b;
            int hi = (a < b) ? b : a;
            unsigned long long key =
                ((unsigned long long)(uint32_t)lo << 32) | (unsigned long long)(uint32_t)hi;
            evid[e] = lookup_edge(hkeys, hval, key);
        }
    }

    uint32_t Ne = counters[0], c1 = counters[1], c2 = counters[2];
    uint32_t Nf = c1 + 2u * c2;
    size_t facesBase = 3u * (size_t)Ne;
    size_t uvIdxBase = facesBase + 3u * (size_t)Nf + 8u * (size_t)NN;

    unsigned long long s = scanArr[t] + blockSums[blockIdx.x];  // global exclusive
    const int* row = &shTbl[(int)c * 6];
    int t4 = t * 4;
    if (ntri == 1) {
        uint32_t fpos = (uint32_t)(s & 0xffffffffull);
        float* f = &out[facesBase + 3u * (size_t)fpos];
        f[0] = (float)evid[row[0]];
        f[1] = (float)evid[row[1]];
        f[2] = (float)evid[row[2]];
        float* u = &out[uvIdxBase + 3u * (size_t)fpos];
        u[0] = (float)t4; u[1] = (float)(t4 + 1); u[2] = (float)(t4 + 2);
    } else {
        uint32_t p2 = (uint32_t)(s >> 32);
        uint32_t fpos = c1 + 2u * p2;
        #pragma unroll
        for (int tri = 0; tri < 2; ++tri) {
            float* f = &out[facesBase + 3u * (size_t)(fpos + (uint32_t)tri)];
            f[0] = (float)evid[row[3 * tri + 0]];
            f[1] = (float)evid[row[3 * tri + 1]];
            f[2] = (float)evid[row[3 * tri + 2]];
            float* u = &out[uvIdxBase + 3u * (size_t)(fpos + (uint32_t)tri)];
            u[0] = (float)t4;
            u[1] = (float)(t4 + tri + 1);
            u[2] = (float)(t4 + tri + 2);
        }
    }
}

// -------------------------------------------------------- kernel: uv lattice
__global__ void k_uvs(int N, int NN, float invN,
                      const uint32_t* __restrict__ counters,
                      float* __restrict__ out) {
    int idx = blockIdx.x * blockDim.x + threadIdx.x;
    int total = NN * 4;
    if (idx >= total) return;
    uint32_t Ne = counters[0], Nf = counters[1] + 2u * counters[2];
    size_t base = 3u * (size_t)Ne + 3u * (size_t)Nf;
    int cell = idx >> 2, p = idx & 3;
    int iy = cell / N, ix = cell - iy * N;
    float x = (float)ix * invN;
    float y = (float)iy * invN;
    float pad = 0.9f * invN;
    float u = x + ((p == 1 || p == 2) ? pad : 0.f);
    float v = y + ((p >= 2) ? pad : 0.f);
    out[base + 2u * (size_t)idx + 0u] = u;
    out[base + 2u * (size_t)idx + 1u] = v;
}

// ---------------------------------------------------------------------------
extern "C" void kernel_launch(void* const* d_in, const int* in_sizes, int n_in,
                              void* d_out, int out_size, void* d_ws, size_t ws_size,
                              hipStream_t stream) {
    const float* pos = (const float*)d_in[0];
    const float* sdf = (const float*)d_in[1];
    const int*   tet = (const int*)d_in[2];
    int NT = in_sizes[2] / 4;

    // N = ceil(sqrt((2*NT+1)//2)) == ceil(sqrt(NT))
    int N = (int)ceil(sqrt((double)NT));
    int NN = N * N;
    float invN = 1.0f / (float)N;

    int nb = (NT + SCAN_BS - 1) / SCAN_BS;

    // workspace carve-up
    char* w = (char*)d_ws;
    unsigned long long* hkeys = (unsigned long long*)w; w += (size_t)HASH_SIZE * 8u;
    uint32_t* hval            = (uint32_t*)w;           w += (size_t)HASH_SIZE * 4u;
    unsigned long long* scanA = (unsigned long long*)w; w += (size_t)NT * 8u;
    uint32_t* code            = (uint32_t*)w;           w += (size_t)NT * 4u;
    unsigned long long* bsums = (unsigned long long*)w; w += (size_t)nb * 8u;
    uint32_t* counters        = (uint32_t*)w;           w += 256u;

    float* out = (float*)d_out;

    k_init<<<dim3((HASH_SIZE / 2u + 255u) / 256u), dim3(256), 0, stream>>>(
        (ulonglong2*)hkeys, counters);
    k_classify_scan1<<<dim3(nb), dim3(SCAN_BS), 0, stream>>>(tet, sdf, NT, code, scanA, bsums);
    k_scan2<<<dim3(1), dim3(1024), 0, stream>>>(bsums, nb, counters);
    k_edges<<<dim3(nb), dim3(SCAN_BS), 0, stream>>>(tet, sdf, pos, NT, code, hkeys, hval,
                                                    counters, out);
    k_faces<<<dim3(nb), dim3(SCAN_BS), 0, stream>>>(tet, NT, NN, code, scanA, bsums,
                                                    hkeys, hval, counters, out);
    k_uvs<<<dim3((NN * 4 + 255) / 256), dim3(256), 0, stream>>>(N, NN, invN, counters, out);
}